// CFiCS_19258633355820
// MI455X (gfx1250) — compile-verified
//
#include <hip/hip_runtime.h>

typedef float v2f __attribute__((ext_vector_type(2)));
typedef float v8f __attribute__((ext_vector_type(8)));

#define N_NODES   50000
#define N_EDGES   800000
#define TEXT_DIM  768
#define GRAPH_DIM 64
#define HIDDEN    128
#define NCLS      14
#define JOINT     (TEXT_DIM + HIDDEN)   // 896

// ---------------------------------------------------------------- utilities

__global__ void zero_kernel(float* __restrict__ p, long n) {
    long i = (long)blockIdx.x * blockDim.x + threadIdx.x;
    if (i < n) p[i] = 0.0f;
}

// pack [Wl1|Wr1] -> W1cat [128,128]; [Wl2|Wr2] -> W2cat [128,256];
// pad Wc [14,896] -> Wcp [16,896]; bc -> bcp[16]
__global__ void prep_weights(const float* __restrict__ Wl1, const float* __restrict__ Wr1,
                             const float* __restrict__ Wl2, const float* __restrict__ Wr2,
                             const float* __restrict__ Wc,  const float* __restrict__ bc,
                             float* __restrict__ W1cat, float* __restrict__ W2cat,
                             float* __restrict__ Wcp,   float* __restrict__ bcp) {
    int i = blockIdx.x * blockDim.x + threadIdx.x;      // 0 .. 32767
    if (i < 128 * 128) {
        int o = i >> 7, j = i & 127;
        W1cat[i] = (j < GRAPH_DIM) ? Wl1[o * GRAPH_DIM + j]
                                   : Wr1[o * GRAPH_DIM + (j - GRAPH_DIM)];
    }
    if (i < 128 * 256) {
        int o = i >> 8, j = i & 255;
        W2cat[i] = (j < HIDDEN) ? Wl2[o * HIDDEN + j]
                                : Wr2[o * HIDDEN + (j - HIDDEN)];
    }
    if (i < 16 * JOINT) {
        int o = i / JOINT, j = i % JOINT;
        Wcp[i] = (o < NCLS) ? Wc[o * JOINT + j] : 0.0f;
    }
    if (i < 16) bcp[i] = (i < NCLS) ? bc[i] : 0.0f;
}

// ---------------------------------------------------------------- scatter-mean

// layer1: gather x_graph[src] (64 feats), scatter-add into X1 cols [0,64)
// (row stride 128); lane j==0 also accumulates the degree.
__global__ void scatter1_kernel(const int* __restrict__ src, const int* __restrict__ dst,
                                const float* __restrict__ xg, float* __restrict__ X1,
                                float* __restrict__ deg) {
    long i = (long)blockIdx.x * blockDim.x + threadIdx.x;   // E * 16
    if (i >= (long)N_EDGES * 16) return;
    int e = (int)(i >> 4), j = (int)(i & 15);
    int s = src[e], d = dst[e];
    float4 v = *(const float4*)(xg + (long)s * GRAPH_DIM + 4 * j);
    float* p = X1 + (long)d * HIDDEN + 4 * j;
    atomicAdd(p + 0, v.x); atomicAdd(p + 1, v.y);
    atomicAdd(p + 2, v.z); atomicAdd(p + 3, v.w);
    if (j == 0) atomicAdd(&deg[d], 1.0f);
}

// X1[:,0:64] /= max(deg,1) ; X1[:,64:128] = x_graph
__global__ void finish1_kernel(const float* __restrict__ deg, const float* __restrict__ xg,
                               float* __restrict__ X1) {
    long i = (long)blockIdx.x * blockDim.x + threadIdx.x;   // N * 64
    if (i >= (long)N_NODES * GRAPH_DIM) return;
    int n = (int)(i >> 6), f = (int)(i & 63);
    float dv = fmaxf(deg[n], 1.0f);
    long r = (long)n * HIDDEN;
    X1[r + f] = X1[r + f] / dv;
    X1[r + GRAPH_DIM + f] = xg[(long)n * GRAPH_DIM + f];
}

// zero X2 cols [0,128) (row stride 256)
__global__ void init2_kernel(float* __restrict__ X2) {
    long i = (long)blockIdx.x * blockDim.x + threadIdx.x;   // N * 128
    if (i >= (long)N_NODES * HIDDEN) return;
    X2[(i >> 7) * 256 + (i & 127)] = 0.0f;
}

// layer2: gather h1[src] (128 feats), scatter-add into X2 cols [0,128) (row stride 256)
__global__ void scatter2_kernel(const int* __restrict__ src, const int* __restrict__ dst,
                                const float* __restrict__ h1, float* __restrict__ X2) {
    long i = (long)blockIdx.x * blockDim.x + threadIdx.x;   // E * 32
    if (i >= (long)N_EDGES * 32) return;
    int e = (int)(i >> 5), j = (int)(i & 31);
    int s = src[e], d = dst[e];
    float4 v = *(const float4*)(h1 + (long)s * HIDDEN + 4 * j);
    float* p = X2 + (long)d * 256 + 4 * j;
    atomicAdd(p + 0, v.x); atomicAdd(p + 1, v.y);
    atomicAdd(p + 2, v.z); atomicAdd(p + 3, v.w);
}

// X2[:,0:128] /= max(deg,1) ; X2[:,128:256] = h1
__global__ void finish2_kernel(const float* __restrict__ deg, const float* __restrict__ h1,
                               float* __restrict__ X2) {
    long i = (long)blockIdx.x * blockDim.x + threadIdx.x;   // N * 128
    if (i >= (long)N_NODES * HIDDEN) return;
    int n = (int)(i >> 7), f = (int)(i & 127);
    float dv = fmaxf(deg[n], 1.0f);
    long r = (long)n * 256;
    X2[r + f] = X2[r + f] / dv;
    X2[r + HIDDEN + f] = h1[(long)n * HIDDEN + f];
}

// ---------------------------------------------------------------- WMMA GEMMs
// V_WMMA_F32_16X16X4_F32 operand layout (wave32):
//   A: lane L holds (row = L%16, K = kk + 2*(L/16), kk+2*(L/16)+1) as float2
//   B: lane L holds (col = L%16, same K pair) as float2  (W row-major [out,K])
//   D: c[v] at lane L -> row v + 8*(L>=16), col L%16
//
// A-register reuse: one wave owns a 16-node stripe, keeps the full A fragment
// (K/2 floats per lane) in VGPRs, and sweeps all NTILES output-column tiles —
// X is read from L2 exactly once instead of NTILES times.

template <int K, int NTILES, bool RELU>
__global__ void sage_gemm_kernel(const float* __restrict__ X, const float* __restrict__ W,
                                 const float* __restrict__ bias, float* __restrict__ out,
                                 int outStride) {
    int lane  = threadIdx.x;            // blockDim = 32
    int tileM = blockIdx.x;
    int rl    = lane & 15;
    int koff  = (lane >> 4) << 1;
    const float* arow = X + (long)(tileM * 16 + rl) * K + koff;

    v2f a[K / 4];
#pragma unroll
    for (int s = 0; s < K / 4; ++s) a[s] = *(const v2f*)(arow + 4 * s);

    int mbase = tileM * 16 + ((lane >> 4) << 3);
#pragma unroll 1
    for (int tn = 0; tn < NTILES; ++tn) {
        const float* brow = W + (long)(tn * 16 + rl) * K + koff;
        v8f c = {0.f, 0.f, 0.f, 0.f, 0.f, 0.f, 0.f, 0.f};
#pragma unroll
        for (int s = 0; s < K / 4; ++s) {
            v2f b = *(const v2f*)(brow + 4 * s);
            c = __builtin_amdgcn_wmma_f32_16x16x4_f32(false, a[s], false, b, (short)0, c,
                                                      false, false);
        }
        int   col = tn * 16 + rl;
        float bv  = bias[col];
#pragma unroll
        for (int v = 0; v < 8; ++v) {
            float val = c[v] + bv;
            if (RELU) val = fmaxf(val, 0.0f);
            out[(long)(mbase + v) * outStride + col] = val;
        }
    }
}

// classifier: K = 896 split across x_text (768, streamed non-temporally) and
// h2 (128, L2-resident); 14 valid output columns.
__global__ void classifier_kernel(const float* __restrict__ xt, const float* __restrict__ h,
                                  const float* __restrict__ Wcp, const float* __restrict__ bcp,
                                  float* __restrict__ logits) {
    int lane  = threadIdx.x;            // blockDim = 32
    int tileM = blockIdx.x;
    int rl    = lane & 15;
    int koff  = (lane >> 4) << 1;
    int node  = tileM * 16 + rl;
    const float* a0   = xt  + (long)node * TEXT_DIM + koff;
    const float* a1   = h   + (long)node * HIDDEN   + koff;
    const float* brow = Wcp + (long)rl   * JOINT    + koff;
    v8f c = {0.f, 0.f, 0.f, 0.f, 0.f, 0.f, 0.f, 0.f};
#pragma unroll 8
    for (int kk = 0; kk < TEXT_DIM; kk += 4) {
        v2f a = __builtin_nontemporal_load((const v2f*)(a0 + kk));  // stream-once
        v2f b = *(const v2f*)(brow + kk);
        c = __builtin_amdgcn_wmma_f32_16x16x4_f32(false, a, false, b, (short)0, c,
                                                  false, false);
    }
#pragma unroll 8
    for (int kk = 0; kk < HIDDEN; kk += 4) {
        v2f a = *(const v2f*)(a1 + kk);
        v2f b = *(const v2f*)(brow + TEXT_DIM + kk);
        c = __builtin_amdgcn_wmma_f32_16x16x4_f32(false, a, false, b, (short)0, c,
                                                  false, false);
    }
    float bv    = bcp[rl];
    int   mbase = tileM * 16 + ((lane >> 4) << 3);
    if (rl < NCLS) {
#pragma unroll
        for (int v = 0; v < 8; ++v)
            __builtin_nontemporal_store(c[v] + bv,
                                        logits + (long)(mbase + v) * NCLS + rl);
    }
}

// ---------------------------------------------------------------- launch

extern "C" void kernel_launch(void* const* d_in, const int* in_sizes, int n_in,
                              void* d_out, int out_size, void* d_ws, size_t ws_size,
                              hipStream_t stream) {
    (void)in_sizes; (void)n_in; (void)out_size; (void)ws_size;
    const float* x_text  = (const float*)d_in[0];
    const float* x_graph = (const float*)d_in[1];
    const int*   edge    = (const int*)d_in[2];
    const int*   src     = edge;
    const int*   dst     = edge + N_EDGES;
    const float* Wl1 = (const float*)d_in[3];
    const float* bl1 = (const float*)d_in[4];
    const float* Wr1 = (const float*)d_in[5];
    const float* Wl2 = (const float*)d_in[6];
    const float* bl2 = (const float*)d_in[7];
    const float* Wr2 = (const float*)d_in[8];
    const float* Wc  = (const float*)d_in[9];
    const float* bc  = (const float*)d_in[10];

    float* logits = (float*)d_out;                              // [N,14]
    float* h2     = (float*)d_out + (long)N_NODES * NCLS;       // [N,128] (tuple's h)

    // workspace layout (floats, all 64B aligned)
    float* ws    = (float*)d_ws;
    float* deg   = ws;                     // 50048
    float* W1cat = deg   + 50048;          // 128*128
    float* W2cat = W1cat + 128 * 128;      // 128*256
    float* Wcp   = W2cat + 128 * 256;      // 16*896
    float* bcp   = Wcp   + 16 * JOINT;     // 32
    float* X2    = bcp   + 32;             // N*256  (X1 aliases same base, N*128)
    float* X1    = X2;
    float* h1    = X2 + (long)N_NODES * 256; // N*128

    const int B = 256;
    // weights prep + zero init
    prep_weights<<<(128 * 256 + B - 1) / B, B, 0, stream>>>(Wl1, Wr1, Wl2, Wr2, Wc, bc,
                                                            W1cat, W2cat, Wcp, bcp);
    zero_kernel<<<(N_NODES + B - 1) / B, B, 0, stream>>>(deg, N_NODES);
    zero_kernel<<<((long)N_NODES * HIDDEN + B - 1) / B, B, 0, stream>>>(X1, (long)N_NODES * HIDDEN);

    // layer-1 scatter-mean (degree folded into scatter, lane j==0)
    scatter1_kernel<<<((long)N_EDGES * 16 + B - 1) / B, B, 0, stream>>>(src, dst, x_graph, X1, deg);
    finish1_kernel<<<((long)N_NODES * GRAPH_DIM + B - 1) / B, B, 0, stream>>>(deg, x_graph, X1);

    // layer-1 GEMM: h1 = relu(X1 @ W1cat^T + bl1)   [K=128, A held in regs, 8 col tiles]
    sage_gemm_kernel<128, HIDDEN / 16, true><<<N_NODES / 16, 32, 0, stream>>>(
        X1, W1cat, bl1, h1, HIDDEN);

    // layer-2 scatter-mean (X2 reuses X1's memory; X1 is dead now)
    init2_kernel<<<((long)N_NODES * HIDDEN + B - 1) / B, B, 0, stream>>>(X2);
    scatter2_kernel<<<((long)N_EDGES * 32 + B - 1) / B, B, 0, stream>>>(src, dst, h1, X2);
    finish2_kernel<<<((long)N_NODES * HIDDEN + B - 1) / B, B, 0, stream>>>(deg, h1, X2);

    // layer-2 GEMM: h2 = relu(X2 @ W2cat^T + bl2)   [K=256] -> straight into d_out
    sage_gemm_kernel<256, HIDDEN / 16, true><<<N_NODES / 16, 32, 0, stream>>>(
        X2, W2cat, bl2, h2, HIDDEN);

    // classifier: logits = [x_text | h2] @ Wc^T + bc
    classifier_kernel<<<N_NODES / 16, 32, 0, stream>>>(x_text, h2, Wcp, bcp, logits);
}